// VQVAE_39633958207792
// MI455X (gfx1250) — compile-verified
//
#include <hip/hip_runtime.h>
#include <hip/hip_bf16.h>

// ---------------------------------------------------------------------------
// VQ-VAE forward on MI455X (gfx1250, wave32).
// GEMM-shaped stages use V_WMMA_F32_16X16X4_F32 (fp32 tensor path on CDNA5).
// Activations are kept channel-last: act[(b*T + t)*C + c].
// dec3 stages weights into LDS with the Tensor Data Mover (tensor_load_to_lds).
// ---------------------------------------------------------------------------

typedef __attribute__((ext_vector_type(2))) float        v2f;
typedef __attribute__((ext_vector_type(8))) float        v8f;
typedef __attribute__((ext_vector_type(4))) unsigned int u32x4;
typedef __attribute__((ext_vector_type(8))) unsigned int u32x8;

#define B_SZ   32
#define L_IN   16384
#define T1     8192      // after enc1
#define T2     4096      // after enc2..enc4 / dec1..dec2
#define DEMB   64
#define KCODES 512
#define NROWS  (B_SZ * T2)      // 131072 position rows
#define LOUT   16385

// ---------------------------------------------------------------------------
// Implicit-im2col fp32 WMMA GEMM:  out[n, m] = relu?(bias[m] + sum_k A[m,k]*B[k,n])
//   rows n = (b, t) positions, k = tap*CIN + ci, m = output channel.
//   t_in = t*STRIDE + tap*DTAP + TOFF   (covers conv and conv-transpose)
//   weight address = m*WS_CO + ci*WS_CI + tap*WS_TAP
// Block = 128 threads = 4 waves; each wave owns one 16-row M tile and NT
// consecutive 16-col N tiles. Per tap, the A fragments (CIN/4 x v2f) are
// preloaded into registers and reused across the NT tiles -> ~1 global b64
// load per WMMA, and NT independent accumulator chains (no WMMA RAW stalls).
// Fragment layout (16x16x4 f32): lanes 0-15 hold K={0,1}, lanes 16-31 K={2,3};
// A lane serves M=lane%16, B lane serves N=lane%16; D: lane = N col, vgpr g =
// row m0 + (lane>=16?8:0) + g  -> 8 contiguous channel-last floats per lane.
// ---------------------------------------------------------------------------
template<int CIN, int COUT, int KT, int STRIDE, int DTAP, int TOFF,
         int WS_CO, int WS_CI, int WS_TAP, bool RELU, int NT>
__global__ __launch_bounds__(128)
void conv_gemm_wmma(const float* __restrict__ act, const float* __restrict__ wgt,
                    const float* __restrict__ bias, float* __restrict__ out,
                    int T_out, int T_in)
{
    const int lane = threadIdx.x & 31;
    const int wv   = threadIdx.x >> 5;
    const int l    = lane & 15;
    const int half = lane >> 4;
    const int koff = half * 2;

    const int ng0 = blockIdx.x * NT;             // first N tile of this block
    const int m0  = (blockIdx.y * 4 + wv) * 16;  // output-channel tile base

    int bArr[NT], tArr[NT];
#pragma unroll
    for (int nt = 0; nt < NT; ++nt) {
        int n = (ng0 + nt) * 16 + l;
        bArr[nt] = n / T_out;
        tArr[nt] = n - bArr[nt] * T_out;
    }

    v8f acc[NT];
#pragma unroll
    for (int nt = 0; nt < NT; ++nt) acc[nt] = (v8f){0.f,0.f,0.f,0.f,0.f,0.f,0.f,0.f};

    for (int tap = 0; tap < KT; ++tap) {
        // ---- preload A fragments for this tap (register-resident) ----
        v2f afr[CIN / 4];
        const float* wrow = wgt + (size_t)(m0 + l) * WS_CO + (size_t)tap * WS_TAP;
#pragma unroll
        for (int s = 0; s < CIN / 4; ++s) {
            const int kk = 4 * s + koff;
            afr[s].x = wrow[(size_t)kk * WS_CI];
            afr[s].y = wrow[(size_t)(kk + 1) * WS_CI];
        }
        // ---- sweep the NT position tiles ----
#pragma unroll
        for (int nt = 0; nt < NT; ++nt) {
            int t_in = tArr[nt] * STRIDE + tap * DTAP + TOFF;
            // branchless zero-pad: clamp address, multiply by validity mask
            float msk = (t_in >= 0 && t_in < T_in) ? 1.0f : 0.0f;
            int tc = t_in < 0 ? 0 : (t_in >= T_in ? T_in - 1 : t_in);
            const float* arow = act + ((size_t)bArr[nt] * T_in + tc) * CIN + koff;
#pragma unroll
            for (int s = 0; s < CIN / 4; ++s) {
                v2f bv = *(const v2f*)(arow + 4 * s);
                bv *= msk;
                acc[nt] = __builtin_amdgcn_wmma_f32_16x16x4_f32(
                              false, afr[s], false, bv, (short)0, acc[nt], false, false);
            }
        }
    }

    const int mb = m0 + half * 8;
    float bv8[8];
#pragma unroll
    for (int g = 0; g < 8; ++g) bv8[g] = bias[mb + g];
#pragma unroll
    for (int nt = 0; nt < NT; ++nt) {
        float* orow = out + ((size_t)bArr[nt] * T_out + tArr[nt]) * COUT + mb;
#pragma unroll
        for (int g = 0; g < 8; ++g) {
            float v = acc[nt][g] + bv8[g];
            if (RELU) v = fmaxf(v, 0.f);
            orow[g] = v;
        }
    }
}

// ---------------------------------------------------------------------------
// enc1: conv1d 1->32, k=4, s=2, p=1 (tiny; VALU). One thread per (b,t).
// ---------------------------------------------------------------------------
__global__ __launch_bounds__(256)
void enc1_k(const float* __restrict__ x, const float* __restrict__ w,
            const float* __restrict__ bias, float* __restrict__ h1)
{
    int g = blockIdx.x * 256 + threadIdx.x;        // n over B*T1
    int b = g >> 13, t = g & (T1 - 1);
    float xv[4];
#pragma unroll
    for (int k = 0; k < 4; ++k) {
        int ti = 2 * t + k - 1;
        xv[k] = (ti >= 0 && ti < L_IN) ? x[(size_t)b * L_IN + ti] : 0.f;
    }
    float* orow = h1 + (size_t)g * 32;
#pragma unroll
    for (int co = 0; co < 32; ++co) {
        float a = bias[co];
#pragma unroll
        for (int k = 0; k < 4; ++k) a = fmaf(w[co * 4 + k], xv[k], a);
        orow[co] = fmaxf(a, 0.f);
    }
}

// ---------------------------------------------------------------------------
// VQ codebook search: scores = ||e||^2 - 2 * z . e  (||z||^2 is row-constant).
// Each wave owns TWO 16-row z tiles (fragments register-resident) and streams
// the 512-code codebook once: every A fragment load feeds two WMMAs on
// independent accumulator chains. Argmin kept per lane, merged across the
// lane^16 pair via wave32 shuffle (first-occurrence tie-break like argmin).
// ---------------------------------------------------------------------------
__global__ __launch_bounds__(128)
void vq_argmin_wmma(const float* __restrict__ z, const float* __restrict__ emb,
                    const float* __restrict__ enorm, int* __restrict__ idx_out,
                    float* __restrict__ idxf_out)
{
    const int lane = threadIdx.x & 31;
    const int wv   = threadIdx.x >> 5;
    const int l    = lane & 15;
    const int half = lane >> 4;
    const int koff = half * 2;
    const int n0   = (blockIdx.x * 4 + wv) * 32 + l;   // rows n0 and n0+16

    v2f zf0[16], zf1[16];                              // two K=64 B-fragments
#pragma unroll
    for (int s = 0; s < 16; ++s) {
        zf0[s] = *(const v2f*)(z + (size_t)n0 * DEMB + 4 * s + koff);
        zf1[s] = *(const v2f*)(z + (size_t)(n0 + 16) * DEMB + 4 * s + koff);
    }

    float best0 = 3.4e38f, best1 = 3.4e38f;
    int   bidx0 = 0,       bidx1 = 0;
    for (int mt = 0; mt < KCODES / 16; ++mt) {
        const int m0 = mt * 16;
        v8f acc0 = (v8f){0.f,0.f,0.f,0.f,0.f,0.f,0.f,0.f};
        v8f acc1 = (v8f){0.f,0.f,0.f,0.f,0.f,0.f,0.f,0.f};
        const float* er = emb + (size_t)(m0 + l) * DEMB + koff;
#pragma unroll
        for (int s = 0; s < 16; ++s) {
            v2f av = *(const v2f*)(er + 4 * s);
            acc0 = __builtin_amdgcn_wmma_f32_16x16x4_f32(
                       false, av, false, zf0[s], (short)0, acc0, false, false);
            acc1 = __builtin_amdgcn_wmma_f32_16x16x4_f32(
                       false, av, false, zf1[s], (short)0, acc1, false, false);
        }
        const int mb = m0 + half * 8;
#pragma unroll
        for (int g = 0; g < 8; ++g) {
            float en = enorm[mb + g];
            float sc0 = en - 2.0f * acc0[g];
            float sc1 = en - 2.0f * acc1[g];
            if (sc0 < best0) { best0 = sc0; bidx0 = mb + g; }
            if (sc1 < best1) { best1 = sc1; bidx1 = mb + g; }
        }
    }
    float ob0 = __shfl_xor(best0, 16, 32);
    int   oi0 = __shfl_xor(bidx0, 16, 32);
    if (ob0 < best0 || (ob0 == best0 && oi0 < bidx0)) { best0 = ob0; bidx0 = oi0; }
    float ob1 = __shfl_xor(best1, 16, 32);
    int   oi1 = __shfl_xor(bidx1, 16, 32);
    if (ob1 < best1 || (ob1 == best1 && oi1 < bidx1)) { best1 = ob1; bidx1 = oi1; }
    if (half == 0) {
        idx_out[n0]       = bidx0;
        idx_out[n0 + 16]  = bidx1;
        idxf_out[n0]      = (float)bidx0;             // indices output as float
        idxf_out[n0 + 16] = (float)bidx1;
    }
}

// ---------------------------------------------------------------------------
// e_norm[k] = ||emb[k]||^2
// ---------------------------------------------------------------------------
__global__ __launch_bounds__(256)
void enorm_k(const float* __restrict__ emb, float* __restrict__ enorm)
{
    int k = blockIdx.x * 256 + threadIdx.x;
    if (k < KCODES) {
        float s = 0.f;
        const float* er = emb + (size_t)k * DEMB;
#pragma unroll
        for (int d = 0; d < DEMB; ++d) s = fmaf(er[d], er[d], s);
        enorm[k] = s;
    }
}

__global__ __launch_bounds__(512)
void zero_k(int* __restrict__ counts, float* __restrict__ loss_acc)
{
    if (threadIdx.x < KCODES) counts[threadIdx.x] = 0;
    if (threadIdx.x == 0)     *loss_acc = 0.f;
}

// ---------------------------------------------------------------------------
// Gather codebook rows -> q, histogram counts, accumulate sum((q-z)^2).
// ---------------------------------------------------------------------------
__global__ __launch_bounds__(256)
void vq_gather_loss(const int* __restrict__ idx, const float* __restrict__ emb,
                    const float* __restrict__ z, float* __restrict__ q,
                    int* __restrict__ counts, float* __restrict__ loss_acc)
{
    int n = blockIdx.x * 256 + threadIdx.x;
    int id = idx[n];
    atomicAdd(&counts[id], 1);
    const float* er = emb + (size_t)id * DEMB;
    const float* zr = z   + (size_t)n  * DEMB;
    float*       qr = q   + (size_t)n  * DEMB;
    float s = 0.f;
#pragma unroll
    for (int d = 0; d < DEMB; ++d) {
        float e  = er[d];
        float dv = e - zr[d];
        qr[d] = e;
        s = fmaf(dv, dv, s);
    }
    __shared__ float red[256];
    red[threadIdx.x] = s;
    __syncthreads();
    for (int off = 128; off > 0; off >>= 1) {
        if (threadIdx.x < off) red[threadIdx.x] += red[threadIdx.x + off];
        __syncthreads();
    }
    if (threadIdx.x == 0) atomicAdd(loss_acc, red[0]);
}

// ---------------------------------------------------------------------------
// loss scalar + perplexity
// ---------------------------------------------------------------------------
__global__ __launch_bounds__(512)
void finalize_k(const int* __restrict__ counts, const float* __restrict__ loss_acc,
                float* __restrict__ out_loss, float* __restrict__ out_perp)
{
    __shared__ float red[512];
    int k = threadIdx.x;
    float p = (float)counts[k] / (float)NROWS;
    red[k] = p * logf(p + 1e-10f);
    __syncthreads();
    for (int off = 256; off > 0; off >>= 1) {
        if (k < off) red[k] += red[k + off];
        __syncthreads();
    }
    if (k == 0) {
        *out_perp = expf(-red[0]);
        *out_loss = 1.25f * (*loss_acc) / ((float)NROWS * (float)DEMB);
    }
}

// ---------------------------------------------------------------------------
// dec3: convT 64->32, k=4, s=2, p=1.  t = 2*t' + k - 1  ->  2 taps per output
// (parity). Weights (64,32,4) = 32 KB staged into LDS by the Tensor Data
// Mover: wave 0 builds a D# (group0: count/lds/global/type, group1: 4-byte
// elements, one 8192-element row) in SGPRs, issues tensor_load_to_lds, waits
// on TENSORcnt, then the workgroup barrier publishes the LDS tile.
// ---------------------------------------------------------------------------
__global__ __launch_bounds__(256)
void dec3_k(const float* __restrict__ d2, const float* __restrict__ w,
            const float* __restrict__ bias, float* __restrict__ d3)
{
    __shared__ float sw[64 * 32 * 4];                  // 32 KB weight tile

    if ((threadIdx.x >> 5) == 0) {                     // wave 0 only (uniform)
        unsigned long long ga = (unsigned long long)(size_t)w;
        unsigned lds = (unsigned)(size_t)(void*)sw;    // LDS byte offset
        u32x4 g0;
        g0[0] = 1u;                                    // count=1, user D#
        g0[1] = lds;                                   // lds_addr
        g0[2] = (unsigned)(ga & 0xFFFFFFFFu);          // global_addr[31:0]
        g0[3] = (unsigned)((ga >> 32) & 0x01FFFFFFu)   // global_addr[56:32]
              | (2u << 30);                            // type=2 ("image")
        u32x8 g1;
        g1[0] = 2u << 16;                              // data_size=4B; mask=0
        g1[1] = 8192u << 16;                           // tensor_dim0[15:0]
        g1[2] = 1u << 16;                              // tensor_dim1 = 1
        g1[3] = 8192u << 16;                           // tile_dim0 = 8192
        g1[4] = 1u;                                    // tile_dim1 = 1
        g1[5] = 8192u;                                 // tensor_dim0_stride
        g1[6] = 0u;
        g1[7] = 0u;
        asm volatile("tensor_load_to_lds %0, %1" :: "s"(g0), "s"(g1) : "memory");
        __builtin_amdgcn_s_wait_tensorcnt(0);
    }
    __syncthreads();

    int g = blockIdx.x * 256 + threadIdx.x;            // n over B*T1
    int b = g >> 13, t = g & (T1 - 1);
    float acc[32];
#pragma unroll
    for (int co = 0; co < 32; ++co) acc[co] = bias[co];

    int k0 = (t + 1) & 1;                              // k parity with (t+1)
#pragma unroll
    for (int kk2 = 0; kk2 < 2; ++kk2) {
        int kk = k0 + 2 * kk2;
        int tp = (t + 1 - kk) >> 1;
        if (tp >= 0 && tp < T2) {
            const float* ar = d2 + ((size_t)b * T2 + tp) * 64;
            for (int ci = 0; ci < 64; ++ci) {
                float a = ar[ci];
                const float* wr = &sw[ci * 128 + kk];
#pragma unroll
                for (int co = 0; co < 32; ++co) acc[co] = fmaf(a, wr[co * 4], acc[co]);
            }
        }
    }
    float* orow = d3 + (size_t)g * 32;
#pragma unroll
    for (int co = 0; co < 32; ++co) orow[co] = fmaxf(acc[co], 0.f);
}

// ---------------------------------------------------------------------------
// dec4: convT 32->1, k=4, s=2, p=1, out_pad=1 -> length 16385.
// ---------------------------------------------------------------------------
__global__ __launch_bounds__(256)
void dec4_k(const float* __restrict__ d3, const float* __restrict__ w,
            const float* __restrict__ bias, float* __restrict__ xr)
{
    int g = blockIdx.x * 256 + threadIdx.x;
    if (g >= B_SZ * LOUT) return;
    int b = g / LOUT, t = g - b * LOUT;
    float acc = bias[0];
    int k0 = (t + 1) & 1;
#pragma unroll
    for (int kk2 = 0; kk2 < 2; ++kk2) {
        int kk = k0 + 2 * kk2;
        int tp = (t + 1 - kk) >> 1;
        if (tp >= 0 && tp < T1) {
            const float* ar = d3 + ((size_t)b * T1 + tp) * 32;
#pragma unroll
            for (int ci = 0; ci < 32; ++ci) acc = fmaf(ar[ci], w[ci * 4 + kk], acc);
        }
    }
    xr[g] = acc;
}

// ---------------------------------------------------------------------------
extern "C" void kernel_launch(void* const* d_in, const int* in_sizes, int n_in,
                              void* d_out, int out_size, void* d_ws, size_t ws_size,
                              hipStream_t stream)
{
    const float* x    = (const float*)d_in[0];
    const float* ew1  = (const float*)d_in[1];
    const float* eb1  = (const float*)d_in[2];
    const float* ew2  = (const float*)d_in[3];
    const float* eb2  = (const float*)d_in[4];
    const float* ew3  = (const float*)d_in[5];
    const float* eb3  = (const float*)d_in[6];
    const float* ew4  = (const float*)d_in[7];
    const float* eb4  = (const float*)d_in[8];
    const float* dw1  = (const float*)d_in[9];
    const float* db1  = (const float*)d_in[10];
    const float* dw2  = (const float*)d_in[11];
    const float* db2  = (const float*)d_in[12];
    const float* dw3  = (const float*)d_in[13];
    const float* db3  = (const float*)d_in[14];
    const float* dw4  = (const float*)d_in[15];
    const float* db4  = (const float*)d_in[16];
    const float* emb  = (const float*)d_in[17];

    float* outf = (float*)d_out;
    // output layout: [0]=vq_loss, [1..524320]=x_recon, [524321]=perplexity,
    //                [524322..]=indices (as float)
    float* out_recon = outf + 1;
    float* out_perp  = outf + 1 + B_SZ * LOUT;
    float* out_idx   = outf + 2 + B_SZ * LOUT;

    char* ws = (char*)d_ws;
    const size_t MB = 1024 * 1024;
    float* R0     = (float*)(ws);                 // 64 MB: h1 / h3 / d1 / d3
    float* R1     = (float*)(ws + 64 * MB);       // 64 MB: h2 / q  / d2
    float* R2     = (float*)(ws + 128 * MB);      // 32 MB: z (lives through VQ)
    float* enorm  = (float*)(ws + 160 * MB);
    int*   counts = (int*)  (ws + 160 * MB + 4096);
    float* lossac = (float*)(ws + 160 * MB + 8192);
    int*   idx    = (int*)  (ws + 160 * MB + 16384);

    zero_k<<<1, 512, 0, stream>>>(counts, lossac);
    enorm_k<<<2, 256, 0, stream>>>(emb, enorm);

    // ---- encoder ----
    enc1_k<<<(B_SZ * T1) / 256, 256, 0, stream>>>(x, ew1, eb1, R0);

    // enc2: 32->64, k4 s2 p1; w (64,32,4): m*128 + ci*4 + tap
    conv_gemm_wmma<32, 64, 4, 2, 1, -1, 128, 4, 1, true, 4>
        <<<dim3(NROWS / 64, 1), 128, 0, stream>>>(R0, ew2, eb2, R1, T2, T1);
    // enc3: 64->128, k3 s1 p1; w (128,64,3): m*192 + ci*3 + tap
    conv_gemm_wmma<64, 128, 3, 1, 1, -1, 192, 3, 1, true, 4>
        <<<dim3(NROWS / 64, 2), 128, 0, stream>>>(R1, ew3, eb3, R0, T2, T2);
    // enc4: 128->64, 1x1; w (64,128,1): m*128 + ci
    conv_gemm_wmma<128, 64, 1, 1, 0, 0, 128, 1, 0, false, 4>
        <<<dim3(NROWS / 64, 1), 128, 0, stream>>>(R0, ew4, eb4, R2, T2, T2);

    // ---- vector quantization ----
    vq_argmin_wmma<<<NROWS / 128, 128, 0, stream>>>(R2, emb, enorm, idx, out_idx);
    vq_gather_loss<<<NROWS / 256, 256, 0, stream>>>(idx, emb, R2, R1, counts, lossac);
    finalize_k<<<1, 512, 0, stream>>>(counts, lossac, outf, out_perp);

    // ---- decoder ----
    // dec1: convT 1x1 64->128; w (64,128,1): addr = ci*128 + m
    conv_gemm_wmma<64, 128, 1, 1, 0, 0, 1, 128, 0, true, 4>
        <<<dim3(NROWS / 64, 2), 128, 0, stream>>>(R1, dw1, db1, R0, T2, T2);
    // dec2: convT k3 s1 p1 128->64; t_in = t - tap + 1; w (128,64,3): ci*192 + m*3 + tap
    conv_gemm_wmma<128, 64, 3, 1, -1, 1, 3, 192, 1, true, 4>
        <<<dim3(NROWS / 64, 1), 128, 0, stream>>>(R0, dw2, db2, R1, T2, T2);
    // dec3: convT k4 s2 p1 64->32 (VALU; TDM-staged LDS weights)
    dec3_k<<<(B_SZ * T1) / 256, 256, 0, stream>>>(R1, dw3, db3, R0);
    // dec4: convT k4 s2 p1 op1 32->1 -> x_recon
    dec4_k<<<(B_SZ * LOUT + 255) / 256, 256, 0, stream>>>(R0, dw4, db4, out_recon);

    (void)in_sizes; (void)n_in; (void)out_size; (void)ws_size;
}